// GraphNetwork_4947802325661
// MI455X (gfx1250) — compile-verified
//
#include <hip/hip_runtime.h>

typedef float v2f __attribute__((ext_vector_type(2)));
typedef float v8f __attribute__((ext_vector_type(8)));

// ---------------------------------------------------------------- utilities
__global__ void zero_f_kernel(float* __restrict__ p, size_t n) {
    size_t i = (size_t)blockIdx.x * blockDim.x + threadIdx.x;
    if (i < n) p[i] = 0.0f;
}

__global__ void zero_i_kernel(int* __restrict__ p, int n) {
    int i = blockIdx.x * blockDim.x + threadIdx.x;
    if (i < n) p[i] = 0;
}

// pad x [N,3] -> xp [N,4] with zero 4th column (power-of-2 rows, no K-tail guards)
__global__ void pad_x_kernel(const float* __restrict__ x, float* __restrict__ xp, int n) {
    int idx = blockIdx.x * blockDim.x + threadIdx.x;   // over N*4
    if (idx < n * 4) {
        int i = idx >> 2, k = idx & 3;
        xp[idx] = (k < 3) ? x[i * 3 + k] : 0.0f;
    }
}

// count in-degree (edges only; self loop added analytically later)
__global__ void degree_kernel(const int* __restrict__ col, int* __restrict__ deg, int nE) {
    int e = blockIdx.x * blockDim.x + threadIdx.x;
    if (e < nE) atomicAdd(&deg[col[e]], 1);
}

// deg_inv[i] = 1 / max(deg_edges[i] + 1, 1)  (self loop => always >= 1)
__global__ void deg_inv_kernel(const int* __restrict__ deg, float* __restrict__ dinv, int n) {
    int i = blockIdx.x * blockDim.x + threadIdx.x;
    if (i < n) dinv[i] = 1.0f / (float)(deg[i] + 1);
}

// -------------------------------------------------------------- edge scatter
// agg[col[e], k] += x[row[e], k]   (unscaled; deg_inv folded into dense pass)

// Generic: one thread per (edge, feature).  Used for DIN < 32 (DIN power of 2).
template <int DIN>
__global__ __launch_bounds__(256) void edge_scatter_kernel(
    const float* __restrict__ x, const int* __restrict__ row,
    const int* __restrict__ col, float* __restrict__ agg, int nE)
{
    long long idx = (long long)blockIdx.x * blockDim.x + threadIdx.x;
    long long total = (long long)nE * DIN;
    if (idx >= total) return;
    int e = (int)(idx / DIN);
    int k = (int)(idx - (long long)e * DIN);
    int r = row[e];
    int c = col[e];
    float v = x[(size_t)r * DIN + k];
    unsafeAtomicAdd(&agg[(size_t)c * DIN + k], v);  // -> global_atomic_add_f32
}

// Wave-per-edge: lanes stripe the feature dim; row/col fetched once per wave.
template <int DIN>
__global__ __launch_bounds__(256) void edge_scatter_wave_kernel(
    const float* __restrict__ x, const int* __restrict__ row,
    const int* __restrict__ col, float* __restrict__ agg, int nE)
{
    int e = (blockIdx.x * 256 + threadIdx.x) >> 5;   // one wave per edge
    if (e >= nE) return;                              // wave-uniform exit
    int lane = threadIdx.x & 31;
    int r = row[e];
    int c = col[e];
    const float* __restrict__ src = x + (size_t)r * DIN;
    float* __restrict__ dst = agg + (size_t)c * DIN;
#pragma unroll
    for (int k = lane; k < DIN; k += 32)
        unsafeAtomicAdd(&dst[k], src[k]);
}

// ------------------------------------------------------------- dense + WMMA
// out[i,:] = leaky( (deg_inv[i]*(agg[i]+x[i])) @ W + x[i] @ R + b )
// Full 16-node tiles; one wave per tile; V_WMMA_F32_16X16X4_F32 chains.
// DSTR = storage row stride of agg/xin (== DIN, except layer 1: DIN=3, DSTR=4;
// padded feature column is zero and weight rows >= DIN are zero in LDS).
template <int DIN, int DOUT, int DSTR>
__global__ __launch_bounds__(256) void dense_wmma_kernel(
    const float* __restrict__ agg, const float* __restrict__ xin,
    const float* __restrict__ deginv, const float* __restrict__ W,
    const float* __restrict__ R, const float* __restrict__ bias,
    float* __restrict__ out, int nTiles)
{
    static_assert(DSTR % 4 == 0 && DSTR >= DIN, "DSTR must be K-padded");
    constexpr int NCT   = (DOUT + 15) / 16;      // 16-wide column tiles
    constexpr int DOUTP = NCT * 16;              // N padded to multiple of 16

    __shared__ float sW[DSTR * DOUTP];
    __shared__ float sR[DSTR * DOUTP];
    __shared__ float sB[DOUTP];
    for (int i = threadIdx.x; i < DSTR * DOUTP; i += 256) {
        int k = i / DOUTP, n = i - k * DOUTP;
        bool ok = (k < DIN) && (n < DOUT);
        sW[i] = ok ? W[k * DOUT + n] : 0.0f;
        sR[i] = ok ? R[k * DOUT + n] : 0.0f;
    }
    for (int i = threadIdx.x; i < DOUTP; i += 256)
        sB[i] = (i < DOUT) ? bias[i] : 0.0f;
    __syncthreads();

    const int wave  = (blockIdx.x * 256 + threadIdx.x) >> 5;  // global wave id
    if (wave >= nTiles) return;                                // wave-uniform exit
    const int lane  = threadIdx.x & 31;
    const int m0    = wave * 16;                               // node tile base

    const int mrow  = lane & 15;   // A-matrix row (M) held by this lane
    const int khalf = lane >> 4;   // 0: K=k0+{0,1} ; 1: K=k0+{2,3}
    const int ncol  = lane & 15;   // B/C column within 16-wide tile
    const int node  = m0 + mrow;
    const float dinv = deginv[node];

    const float* __restrict__ aRow = agg + (size_t)node * DSTR;
    const float* __restrict__ xRow = xin + (size_t)node * DSTR;

    v8f acc[NCT] = {};

#pragma unroll
    for (int k0 = 0; k0 < DSTR; k0 += 4) {
        const int ka = k0 + 2 * khalf;
        float x0 = xRow[ka + 0], x1 = xRow[ka + 1];   // unconditional (padded)
        float g0 = aRow[ka + 0], g1 = aRow[ka + 1];
        v2f aTot, aX;
        aTot[0] = dinv * (g0 + x0); aTot[1] = dinv * (g1 + x1);  // (agg+self)*deg_inv
        aX[0] = x0; aX[1] = x1;                                  // root path
#pragma unroll
        for (int ct = 0; ct < NCT; ++ct) {
            const int n = ct * 16 + ncol;
            v2f bW, bR;                       // padded LDS -> unconditional loads
            bW[0] = sW[(ka + 0) * DOUTP + n];
            bW[1] = sW[(ka + 1) * DOUTP + n];
            bR[0] = sR[(ka + 0) * DOUTP + n];
            bR[1] = sR[(ka + 1) * DOUTP + n];
            acc[ct] = __builtin_amdgcn_wmma_f32_16x16x4_f32(
                false, aTot, false, bW, (short)0, acc[ct], false, false);
            acc[ct] = __builtin_amdgcn_wmma_f32_16x16x4_f32(
                false, aX, false, bR, (short)0, acc[ct], false, false);
        }
    }

    // C layout: VGPR r -> lanes 0-15: (M=r, N=lane) ; lanes 16-31: (M=r+8, N=lane-16)
#pragma unroll
    for (int ct = 0; ct < NCT; ++ct) {
        const int n = ct * 16 + ncol;
        if (DOUTP == DOUT || n < DOUT) {      // compile-time true for DOUT%16==0
            const float bv = sB[n];
#pragma unroll
            for (int r = 0; r < 8; ++r) {
                const int ri = m0 + r + 8 * khalf;   // always in range (full tile)
                float v = acc[ct][r] + bv;
                v = fmaxf(v, 0.1f * v);              // leaky_relu(0.1) as max
                out[(size_t)ri * DOUT + n] = v;
            }
        }
    }
}

// Scalar tail for the (N % 16) leftover nodes; not launched when remainder == 0.
template <int DIN, int DOUT, int DSTR>
__global__ void dense_tail_kernel(
    const float* __restrict__ agg, const float* __restrict__ xin,
    const float* __restrict__ deginv, const float* __restrict__ W,
    const float* __restrict__ R, const float* __restrict__ bias,
    float* __restrict__ out, int start, int nNodes)
{
    int i = start + (int)blockIdx.x;
    int n = threadIdx.x;
    if (i >= nNodes || n >= DOUT) return;
    float di = deginv[i];
    float s = bias[n];
    for (int k = 0; k < DIN; ++k) {
        float xv = xin[(size_t)i * DSTR + k];
        float at = di * (agg[(size_t)i * DSTR + k] + xv);
        s += at * W[k * DOUT + n] + xv * R[k * DOUT + n];
    }
    s = fmaxf(s, 0.1f * s);
    out[(size_t)i * DOUT + n] = s;
}

// ----------------------------------------------------------------- per-layer
// DSTR: storage stride of xin/agg rows (padded K).
template <int DIN, int DOUT, int DSTR = DIN>
static void run_layer(const float* xin, float* xout,
                      const float* W, const float* b, const float* R,
                      const int* row, const int* col, const float* dinv,
                      float* agg, int N, int E, hipStream_t stream)
{
    size_t aggN = (size_t)N * DSTR;
    zero_f_kernel<<<(unsigned)((aggN + 511) / 512), 512, 0, stream>>>(agg, aggN);

    if constexpr (DSTR >= 32) {
        unsigned eblocks = (unsigned)(((long long)E * 32 + 255) / 256);
        edge_scatter_wave_kernel<DSTR><<<eblocks, 256, 0, stream>>>(xin, row, col, agg, E);
    } else {
        long long ework = (long long)E * DSTR;
        unsigned eblocks = (unsigned)((ework + 255) / 256);
        edge_scatter_kernel<DSTR><<<eblocks, 256, 0, stream>>>(xin, row, col, agg, E);
    }

    int tiles = N / 16;                         // full 16-node tiles (WMMA path)
    if (tiles > 0) {
        unsigned dblocks = (unsigned)(((long long)tiles * 32 + 255) / 256);
        dense_wmma_kernel<DIN, DOUT, DSTR><<<dblocks, 256, 0, stream>>>(
            agg, xin, dinv, W, R, b, xout, tiles);
    }
    int rem = N - tiles * 16;                   // scalar tail (0 for N=100000)
    if (rem > 0) {
        dense_tail_kernel<DIN, DOUT, DSTR><<<rem, DOUT, 0, stream>>>(
            agg, xin, dinv, W, R, b, xout, tiles * 16, N);
    }
}

extern "C" void kernel_launch(void* const* d_in, const int* in_sizes, int n_in,
                              void* d_out, int out_size, void* d_ws, size_t ws_size,
                              hipStream_t stream)
{
    const float* x0  = (const float*)d_in[0];
    const int*   ei  = (const int*)d_in[1];
    const int N = in_sizes[0] / 3;
    const int E = in_sizes[1] / 2;
    const int* row = ei;          // edge_index[0,:]
    const int* col = ei + E;      // edge_index[1,:]

    // workspace (floats): [deg:int N][deg_inv N][agg N*64][buf0 N*64][buf1 N*64][xp N*4]
    float* ws   = (float*)d_ws;
    int*   deg  = (int*)ws;
    float* dinv = ws + N;
    float* agg  = ws + 2 * (size_t)N;
    float* buf0 = agg + (size_t)N * 64;
    float* buf1 = buf0 + (size_t)N * 64;
    float* xp   = buf1 + (size_t)N * 64;

    // degree -> deg_inv (recomputed every call; deterministic)
    zero_i_kernel<<<(N + 255) / 256, 256, 0, stream>>>(deg, N);
    degree_kernel<<<(E + 255) / 256, 256, 0, stream>>>(col, deg, E);
    deg_inv_kernel<<<(N + 255) / 256, 256, 0, stream>>>(deg, dinv, N);

    // pad input features to stride 4 (zero 4th column)
    pad_x_kernel<<<(N * 4 + 255) / 256, 256, 0, stream>>>(x0, xp, N);

    const float* W[8]; const float* B[8]; const float* Rm[8];
    for (int l = 0; l < 8; ++l) {
        W[l]  = (const float*)d_in[2 + 3 * l];
        B[l]  = (const float*)d_in[3 + 3 * l];
        Rm[l] = (const float*)d_in[4 + 3 * l];
    }
    float* outF = (float*)d_out;

    run_layer< 3, 16, 4>(xp,   buf0, W[0], B[0], Rm[0], row, col, dinv, agg, N, E, stream);
    run_layer<16, 16>(buf0, buf1, W[1], B[1], Rm[1], row, col, dinv, agg, N, E, stream);
    run_layer<16, 32>(buf1, buf0, W[2], B[2], Rm[2], row, col, dinv, agg, N, E, stream);
    run_layer<32, 64>(buf0, buf1, W[3], B[3], Rm[3], row, col, dinv, agg, N, E, stream);
    run_layer<64, 64>(buf1, buf0, W[4], B[4], Rm[4], row, col, dinv, agg, N, E, stream);
    run_layer<64, 32>(buf0, buf1, W[5], B[5], Rm[5], row, col, dinv, agg, N, E, stream);
    run_layer<32, 16>(buf1, buf0, W[6], B[6], Rm[6], row, col, dinv, agg, N, E, stream);
    run_layer<16,  1>(buf0, outF, W[7], B[7], Rm[7], row, col, dinv, agg, N, E, stream);
}